// MOE_66056597012811
// MI455X (gfx1250) — compile-verified
//
#include <hip/hip_runtime.h>

#define N_EXPERTS 8
#define N_EMBED   1024
#define BB        4
#define TT        1024
#define NTOK      (BB * TT)       // 4096 tokens
#define TOPK      2
#define CAP       NTOK            // max rows per expert (top-2 experts are distinct)
#define MT_MAX    (CAP / 16)      // 256
#define NT4       (N_EMBED / 64)  // 16 column-tiles of width 64 per wave

typedef __attribute__((ext_vector_type(2))) float v2f;
typedef __attribute__((ext_vector_type(8))) float v8f;

// ---------------------------------------------------------------------------
// Kernel 0: zero the y output region and the per-expert counters.
// ---------------------------------------------------------------------------
__global__ void moe_zero(float* __restrict__ y, int* __restrict__ counts) {
    int i = blockIdx.x * blockDim.x + threadIdx.x;
    const int n = NTOK * N_EMBED;
    for (int j = i; j < n; j += gridDim.x * blockDim.x) y[j] = 0.0f;
    if (i < N_EXPERTS) counts[i] = 0;
}

// ---------------------------------------------------------------------------
// Kernel 1: gating. One wave32 per token: logits = x @ Wg + b, top-2 (jax
// tie-break: lowest index wins on ties), 2-way softmax, write probs, append
// assignment ids (token*2+k) to per-expert row lists.
// ---------------------------------------------------------------------------
__global__ void moe_gate(const float* __restrict__ x,
                         const float* __restrict__ wg,
                         const float* __restrict__ bias,
                         float* __restrict__ probs_out,
                         int* __restrict__ counts,
                         int* __restrict__ rowmap) {
    const int lane = threadIdx.x & 31;
    const int tok  = blockIdx.x * (blockDim.x >> 5) + (threadIdx.x >> 5);
    if (tok >= NTOK) return;

    const float* __restrict__ xr = x + (size_t)tok * N_EMBED;

    float acc[N_EXPERTS];
#pragma unroll
    for (int e = 0; e < N_EXPERTS; ++e) acc[e] = 0.0f;

#pragma unroll 4
    for (int i = 0; i < N_EMBED / 32; ++i) {
        const int c = lane + 32 * i;
        const float xv = xr[c];
        const float4* w4 = (const float4*)(wg + (size_t)c * N_EXPERTS);
        const float4 wa = w4[0];
        const float4 wb = w4[1];
        acc[0] += xv * wa.x; acc[1] += xv * wa.y;
        acc[2] += xv * wa.z; acc[3] += xv * wa.w;
        acc[4] += xv * wb.x; acc[5] += xv * wb.y;
        acc[6] += xv * wb.z; acc[7] += xv * wb.w;
    }

    // wave32 all-reduce for each expert logit
#pragma unroll
    for (int off = 16; off > 0; off >>= 1) {
#pragma unroll
        for (int e = 0; e < N_EXPERTS; ++e)
            acc[e] += __shfl_xor(acc[e], off, 32);
    }

    if (lane == 0) {
        float l[N_EXPERTS];
#pragma unroll
        for (int e = 0; e < N_EXPERTS; ++e) l[e] = acc[e] + bias[e];

        // top-1 (strict > keeps lowest index on ties, matching lax.top_k)
        int i1 = 0;
#pragma unroll
        for (int e = 1; e < N_EXPERTS; ++e) if (l[e] > l[i1]) i1 = e;
        // top-2
        int i2 = (i1 == 0) ? 1 : 0;
#pragma unroll
        for (int e = 0; e < N_EXPERTS; ++e)
            if (e != i1 && l[e] > l[i2]) i2 = e;

        const float eb = __expf(l[i2] - l[i1]);   // <= 1
        const float inv = 1.0f / (1.0f + eb);
        probs_out[tok * 2 + 0] = inv;
        probs_out[tok * 2 + 1] = eb * inv;

        const int s1 = atomicAdd(&counts[i1], 1);
        rowmap[i1 * CAP + s1] = tok * 2 + 0;
        const int s2 = atomicAdd(&counts[i2], 1);
        rowmap[i2 * CAP + s2] = tok * 2 + 1;
    }
}

// ---------------------------------------------------------------------------
// Kernel 2: grouped GEMM via V_WMMA_F32_16X16X4_F32, 16x64 tile per wave.
// One A fragment feeds 4 accumulators (N-subtiles n, n+16, n+32, n+48),
// cutting per-WMMA load bytes from 16B to 10B. Grid = E * MT_MAX * NT4;
// tiles beyond count[e] exit early (uniform branch => EXEC all-ones).
//
// Fragment layouts (ISA 7.12.2, wave32):
//   A 16x4 : lane m = lane&15 ; VGPR0/1 hold K = koff, koff+1, koff = 2*(lane>>4)
//   B 4x16 : lane n = lane&15 ; VGPR0/1 hold K = koff, koff+1
//   C 16x16: lane n = lane&15 ; VGPR r holds row M = r + 8*(lane>>4)
// ---------------------------------------------------------------------------
__global__ void moe_gemm(const float* __restrict__ x,
                         const float* __restrict__ w,       // (E, C, C) row-major
                         const float* __restrict__ probs,   // (NTOK, 2)
                         const int* __restrict__ counts,
                         const int* __restrict__ rowmap,
                         float* __restrict__ y) {
    const int bid = blockIdx.x;
    const int e   = bid / (MT_MAX * NT4);
    const int rem = bid % (MT_MAX * NT4);
    const int mt  = rem / NT4;
    const int nt  = rem % NT4;

    const int cnt = counts[e];
    if (mt * 16 >= cnt) return;

    const int lane = threadIdx.x;      // block = 1 wave32
    const int half = lane >> 4;        // 0 or 1
    const int l16  = lane & 15;
    const int koff = half * 2;

    // A row gathered straight from x via the expert row list.
    const int rg  = mt * 16 + l16;
    const int rc  = rg < cnt ? rg : cnt - 1;
    const int aid = rowmap[e * CAP + rc];
    const float* __restrict__ arow =
        x + (size_t)(aid >> 1) * N_EMBED + koff;

    // B base: row koff of W_e, columns [nt*64 + l16 + 16*j], j = 0..3
    const float* __restrict__ bbase =
        w + (size_t)e * N_EMBED * N_EMBED + (size_t)koff * N_EMBED + nt * 64 + l16;

    v8f acc0 = {}, acc1 = {}, acc2 = {}, acc3 = {};

#pragma unroll 2
    for (int k = 0; k < N_EMBED; k += 4) {
        v2f a = *(const v2f*)(arow + k);

        const float* b0 = bbase + (size_t)k * N_EMBED;   // row k+koff
        const float* b1 = b0 + N_EMBED;                  // row k+koff+1

        v2f bv0, bv1, bv2, bv3;
        bv0.x = b0[0];  bv0.y = b1[0];
        bv1.x = b0[16]; bv1.y = b1[16];
        bv2.x = b0[32]; bv2.y = b1[32];
        bv3.x = b0[48]; bv3.y = b1[48];

        acc0 = __builtin_amdgcn_wmma_f32_16x16x4_f32(false, a, false, bv0,
                                                     (short)0, acc0, false, false);
        acc1 = __builtin_amdgcn_wmma_f32_16x16x4_f32(false, a, false, bv1,
                                                     (short)0, acc1, false, false);
        acc2 = __builtin_amdgcn_wmma_f32_16x16x4_f32(false, a, false, bv2,
                                                     (short)0, acc2, false, false);
        acc3 = __builtin_amdgcn_wmma_f32_16x16x4_f32(false, a, false, bv3,
                                                     (short)0, acc3, false, false);
    }

    // Scatter D, scaled by routing prob. Each y element gets exactly two adds
    // total (one per routed expert) -> commutative, bit-deterministic.
#pragma unroll
    for (int r = 0; r < 8; ++r) {
        const int g = mt * 16 + r + 8 * half;
        if (g < cnt) {
            const int a2  = rowmap[e * CAP + g];
            const float p = probs[a2];
            float* __restrict__ yr =
                y + (size_t)(a2 >> 1) * N_EMBED + nt * 64 + l16;
            atomicAdd(yr +  0, acc0[r] * p);
            atomicAdd(yr + 16, acc1[r] * p);
            atomicAdd(yr + 32, acc2[r] * p);
            atomicAdd(yr + 48, acc3[r] * p);
        }
    }
}

// ---------------------------------------------------------------------------
extern "C" void kernel_launch(void* const* d_in, const int* in_sizes, int n_in,
                              void* d_out, int out_size, void* d_ws, size_t ws_size,
                              hipStream_t stream) {
    (void)in_sizes; (void)n_in; (void)out_size; (void)ws_size;

    const float* x    = (const float*)d_in[0];   // (B, T, C) f32
    const float* wg   = (const float*)d_in[1];   // (C, E)    f32
    const float* bias = (const float*)d_in[2];   // (E,)      f32
    const float* wfc  = (const float*)d_in[3];   // (E, C, C) f32

    float* y     = (float*)d_out;                         // (B, T, C)
    float* probs = y + (size_t)NTOK * N_EMBED;            // (B, T, 2)

    int* counts = (int*)d_ws;          // 8 ints (padded to 16)
    int* rowmap = counts + 16;         // 8 * 4096 ints

    moe_zero<<<2048, 256, 0, stream>>>(y, counts);
    moe_gate<<<NTOK / 8, 256, 0, stream>>>(x, wg, bias, probs, counts, rowmap);
    moe_gemm<<<N_EXPERTS * MT_MAX * NT4, 32, 0, stream>>>(x, wfc, probs, counts,
                                                          rowmap, y);
}